// PNA_26207890440680
// MI455X (gfx1250) — compile-verified
//
#include <hip/hip_runtime.h>
#include <cmath>

#define FDIM 64

typedef float v2f __attribute__((ext_vector_type(2)));
typedef float v8f __attribute__((ext_vector_type(8)));

__device__ __forceinline__ v8f wmma4(v2f a, v2f b, v8f c) {
    // D = A(16x4 f32) * B(4x16 f32) + C(16x16 f32)
    return __builtin_amdgcn_wmma_f32_16x16x4_f32(
        false, a, false, b, (short)0, c, false, false);
}

// order-preserving float<->uint for native u32 atomic max/min
__device__ __forceinline__ unsigned f2ord(float f) {
    unsigned u = __float_as_uint(f);
    return (u & 0x80000000u) ? ~u : (u | 0x80000000u);
}
__device__ __forceinline__ float ord2f(unsigned o) {
    unsigned u = (o & 0x80000000u) ? (o & 0x7FFFFFFFu) : ~o;
    return __uint_as_float(u);
}

// ---------------- degree precompute ----------------
__global__ void k_deg_init(float* degf, float* cntf, int N) {
    int i = blockIdx.x * blockDim.x + threadIdx.x;
    if (i < N) { degf[i] = 0.f; cntf[i] = 0.f; }
}

__global__ void k_deg_count(const int* __restrict__ ei, int E,
                            float* degf, float* cntf) {
    int e = blockIdx.x * blockDim.x + threadIdx.x;
    if (e >= E) return;
    atomicAdd(&degf[ei[e]], 1.f);        // src row
    atomicAdd(&cntf[ei[E + e]], 1.f);    // dst row
}

__global__ void k_scalars(const float* __restrict__ degf, const float* __restrict__ cntf,
                          float* amp, float* att, float* inv_safe, float* has,
                          int N, float avg_log) {
    int i = blockIdx.x * blockDim.x + threadIdx.x;
    if (i >= N) return;
    float dl = logf(fmaxf(degf[i], 1.f) + 1.f);
    amp[i] = dl / avg_log;
    att[i] = avg_log / dl;
    float c = cntf[i];
    inv_safe[i] = 1.f / fmaxf(c, 1.f);
    has[i] = (c > 0.f) ? 1.f : 0.f;
}

// ---------------- B = h @ Wbot   (+ init stats) ----------------
__global__ void __launch_bounds__(128)
k_transformB(const float* __restrict__ h, const float* __restrict__ Wb,
             float* __restrict__ Bm, float* __restrict__ sumB, float* __restrict__ sumB2,
             unsigned* __restrict__ maxO, unsigned* __restrict__ minO, int N) {
    __shared__ float hs[16][67];
    int tid = threadIdx.x;
    int rb = blockIdx.x * 16;

    for (int i = tid; i < 16 * FDIM; i += 128) {
        int r = i >> 6, c = i & 63;
        int node = min(rb + r, N - 1);
        hs[r][c] = h[node * FDIM + c];
    }
    for (int i = tid; i < 16 * FDIM; i += 128) {
        int r = i >> 6, c = i & 63;
        int node = rb + r;
        if (node < N) {
            int gi = node * FDIM + c;
            sumB[gi] = 0.f; sumB2[gi] = 0.f;
            maxO[gi] = 0u;  minO[gi] = 0xFFFFFFFFu;
        }
    }
    __syncthreads();

    int wave = tid >> 5, lane = tid & 31;
    int hi = lane >> 4, ln = lane & 15;
    int cb = wave * 16;

    v8f acc = {};
#pragma unroll
    for (int kk = 0; kk < 16; ++kk) {
        int k = kk * 4 + hi * 2;
        v2f a, b;
        a.x = hs[ln][k];     a.y = hs[ln][k + 1];
        b.x = Wb[k * FDIM + cb + ln];
        b.y = Wb[(k + 1) * FDIM + cb + ln];
        acc = wmma4(a, b, acc);
    }
#pragma unroll
    for (int r = 0; r < 8; ++r) {
        int node = rb + r + 8 * hi;
        if (node < N) Bm[node * FDIM + cb + ln] = acc[r];
    }
}

// ---------------- per-edge stats: 16 lanes/edge, float4/lane ----------------
__global__ void __launch_bounds__(256)
k_edge(const int* __restrict__ ei, int E, const float* __restrict__ Bm,
       float* __restrict__ sumB, float* __restrict__ sumB2,
       unsigned* __restrict__ maxO, unsigned* __restrict__ minO) {
    int t = blockIdx.x * blockDim.x + threadIdx.x;
    int e = t >> 4;
    if (e >= E) return;
    int f4 = (t & 15) * 4;
    int src = ei[e];
    int dst = ei[E + e];
    float4 v = *(const float4*)(Bm + src * FDIM + f4);
    float* sb = sumB + dst * FDIM + f4;
    float* s2 = sumB2 + dst * FDIM + f4;
    unsigned* mo = maxO + dst * FDIM + f4;
    unsigned* mi = minO + dst * FDIM + f4;
    float vv[4] = {v.x, v.y, v.z, v.w};
#pragma unroll
    for (int j = 0; j < 4; ++j) {
        atomicAdd(sb + j, vv[j]);
        atomicAdd(s2 + j, vv[j] * vv[j]);
        unsigned o = f2ord(vv[j]);
        atomicMax(mo + j, o);
        atomicMin(mi + j, o);
    }
}

// ---------------- node update: A=h@Wtop, aggregate, post_nn, lin+residual ----
__global__ void __launch_bounds__(128)
k_update(const float* __restrict__ h, float* __restrict__ hout,
         const float* __restrict__ Wtop, const float* __restrict__ preB,
         const float* __restrict__ postW, const float* __restrict__ postB,
         const float* __restrict__ linW, const float* __restrict__ linB,
         const float* __restrict__ sumB, const float* __restrict__ sumB2,
         const unsigned* __restrict__ maxO, const unsigned* __restrict__ minO,
         const float* __restrict__ cntf, const float* __restrict__ inv_safe,
         const float* __restrict__ hasv, const float* __restrict__ ampv,
         const float* __restrict__ attv, int N) {
    __shared__ float xcat[16][835];   // [h(64) | mean mx mn std (256) | *amp(256) | *att(256)]
    __shared__ float upd[16][67];
    __shared__ float s_cnt[16], s_inv[16], s_has[16], s_amp[16], s_att[16];

    int tid = threadIdx.x;
    int rb = blockIdx.x * 16;

    for (int i = tid; i < 16 * FDIM; i += 128) {
        int r = i >> 6, c = i & 63;
        int node = min(rb + r, N - 1);
        xcat[r][c] = h[node * FDIM + c];
    }
    if (tid < 16) {
        int node = min(rb + tid, N - 1);
        s_cnt[tid] = cntf[node];   s_inv[tid] = inv_safe[node];
        s_has[tid] = hasv[node];   s_amp[tid] = ampv[node];
        s_att[tid] = attv[node];
    }
    __syncthreads();

    int wave = tid >> 5, lane = tid & 31;
    int hi = lane >> 4, ln = lane & 15;
    int cb = wave * 16;
    int col = cb + ln;

    // ---- GEMM1: A-tile = h_tile @ Wtop  (K=64) ----
    v8f accA = {};
#pragma unroll
    for (int kk = 0; kk < 16; ++kk) {
        int k = kk * 4 + hi * 2;
        v2f a, b;
        a.x = xcat[ln][k];   a.y = xcat[ln][k + 1];
        b.x = Wtop[k * FDIM + col];
        b.y = Wtop[(k + 1) * FDIM + col];
        accA = wmma4(a, b, accA);
    }

    // ---- aggregate stats per owned element, build xcat cols >= 64 ----
    float pb = preB[col];
#pragma unroll
    for (int r = 0; r < 8; ++r) {
        int row = r + 8 * hi;
        int node = min(rb + row, N - 1);
        int gi = node * FDIM + col;
        float c_ = accA[r] + pb;
        float cnt = s_cnt[row], inv = s_inv[row], hs_ = s_has[row];
        float sb = sumB[gi], s2 = sumB2[gi];
        float mxb = ord2f(maxO[gi]), mnb = ord2f(minO[gi]);
        float mean = (cnt * c_ + sb) * inv;
        float msq  = (cnt * c_ * c_ + 2.f * c_ * sb + s2) * inv;
        float var = msq - mean * mean;
        var = var > 0.f ? var : 0.f;
        float sd = sqrtf(var + 1e-5f);
        float vmx = (hs_ != 0.f) ? (c_ + mxb) : 0.f;
        float vmn = (hs_ != 0.f) ? (c_ + mnb) : 0.f;
        float am = s_amp[row], at = s_att[row];
        xcat[row][ 64 + col] = mean;      xcat[row][128 + col] = vmx;
        xcat[row][192 + col] = vmn;       xcat[row][256 + col] = sd;
        xcat[row][320 + col] = mean * am; xcat[row][384 + col] = vmx * am;
        xcat[row][448 + col] = vmn * am;  xcat[row][512 + col] = sd * am;
        xcat[row][576 + col] = mean * at; xcat[row][640 + col] = vmx * at;
        xcat[row][704 + col] = vmn * at;  xcat[row][768 + col] = sd * at;
    }
    __syncthreads();

    // ---- GEMM2: upd = xcat(16x832) @ postW(832x64)  (K=832) ----
    v8f accU = {};
#pragma unroll 4
    for (int kk = 0; kk < 208; ++kk) {
        int k = kk * 4 + hi * 2;
        v2f a, b;
        a.x = xcat[ln][k];   a.y = xcat[ln][k + 1];
        b.x = postW[k * FDIM + col];
        b.y = postW[(k + 1) * FDIM + col];
        accU = wmma4(a, b, accU);
    }
    float pob = postB[col];
#pragma unroll
    for (int r = 0; r < 8; ++r)
        upd[r + 8 * hi][col] = accU[r] + pob;
    __syncthreads();

    // ---- GEMM3: hnew = upd @ linW + lin_b + h  (K=64) ----
    v8f accO = {};
#pragma unroll
    for (int kk = 0; kk < 16; ++kk) {
        int k = kk * 4 + hi * 2;
        v2f a, b;
        a.x = upd[ln][k];   a.y = upd[ln][k + 1];
        b.x = linW[k * FDIM + col];
        b.y = linW[(k + 1) * FDIM + col];
        accO = wmma4(a, b, accO);
    }
    float lb = linB[col];
#pragma unroll
    for (int r = 0; r < 8; ++r) {
        int row = r + 8 * hi;
        int node = rb + row;
        if (node < N)
            hout[node * FDIM + col] = accO[r] + lb + xcat[row][col];
    }
}

// ---------------------------------------------------------------------------
extern "C" void kernel_launch(void* const* d_in, const int* in_sizes, int n_in,
                              void* d_out, int out_size, void* d_ws, size_t ws_size,
                              hipStream_t stream) {
    const float* x     = (const float*)d_in[0];
    const int*   ei    = (const int*)d_in[1];
    const float* preW  = (const float*)d_in[2];   // [L,128,64]
    const float* preBb = (const float*)d_in[3];   // [L,64]
    const float* postW = (const float*)d_in[4];   // [L,832,64]
    const float* postB = (const float*)d_in[5];   // [L,64]
    const float* linW  = (const float*)d_in[6];   // [L,64,64]
    const float* linB  = (const float*)d_in[7];   // [L,64]
    float* out = (float*)d_out;

    const int N = in_sizes[0] / FDIM;
    const int E = in_sizes[1] / 2;
    const int L = in_sizes[2] / (2 * FDIM * FDIM);

    // AVG_LOG: hist = [1]*10 + [20]*10
    double s = 0.0, w = 0.0;
    for (int b = 0; b < 20; ++b) {
        double hc = (b < 10) ? 1.0 : 20.0;
        s += std::log((double)b + 1.0) * hc;
        w += hc;
    }
    float avg_log = (float)(s / w);

    // workspace partition
    char* p = (char*)d_ws;
    size_t nf = (size_t)N * FDIM;
    float*    Bm    = (float*)p;    p += nf * 4;
    float*    sumB  = (float*)p;    p += nf * 4;
    float*    sumB2 = (float*)p;    p += nf * 4;
    unsigned* maxO  = (unsigned*)p; p += nf * 4;
    unsigned* minO  = (unsigned*)p; p += nf * 4;
    float*    hA    = (float*)p;    p += nf * 4;
    float*    degf  = (float*)p;    p += (size_t)N * 4;
    float*    cntf  = (float*)p;    p += (size_t)N * 4;
    float*    ampv  = (float*)p;    p += (size_t)N * 4;
    float*    attv  = (float*)p;    p += (size_t)N * 4;
    float*    invs  = (float*)p;    p += (size_t)N * 4;
    float*    hasv  = (float*)p;    p += (size_t)N * 4;

    int tiles = (N + 15) / 16;

    k_deg_init<<<(N + 255) / 256, 256, 0, stream>>>(degf, cntf, N);
    k_deg_count<<<(E + 255) / 256, 256, 0, stream>>>(ei, E, degf, cntf);
    k_scalars<<<(N + 255) / 256, 256, 0, stream>>>(degf, cntf, ampv, attv, invs, hasv, N, avg_log);

    for (int l = 0; l < L; ++l) {
        const float* hin  = (l == 0) ? x : ((l & 1) ? out : hA);
        float*       hout = (l & 1) ? hA : out;   // l=0 -> out, l=1 -> hA, l=2 -> out
        const float* Wtop = preW + (size_t)l * 2 * FDIM * FDIM;
        const float* Wbot = Wtop + (size_t)FDIM * FDIM;

        k_transformB<<<tiles, 128, 0, stream>>>(hin, Wbot, Bm, sumB, sumB2, maxO, minO, N);
        k_edge<<<(E * 16 + 255) / 256, 256, 0, stream>>>(ei, E, Bm, sumB, sumB2, maxO, minO);
        k_update<<<tiles, 128, 0, stream>>>(hin, hout,
                                            Wtop, preBb + (size_t)l * FDIM,
                                            postW + (size_t)l * 13 * FDIM * FDIM,
                                            postB + (size_t)l * FDIM,
                                            linW + (size_t)l * FDIM * FDIM,
                                            linB + (size_t)l * FDIM,
                                            sumB, sumB2, maxO, minO,
                                            cntf, invs, hasv, ampv, attv, N);
    }
}